// EdgeTypeRGCN_79637283602842
// MI455X (gfx1250) — compile-verified
//
#include <hip/hip_runtime.h>
#include <math.h>

typedef float v2f __attribute__((ext_vector_type(2)));
typedef float v8f __attribute__((ext_vector_type(8)));

static constexpr int kD       = 256;
static constexpr int kNB      = 4;
static constexpr int kNodes   = 20000;
static constexpr int kEdges   = 640000;
static constexpr int kHbCols  = kNB * kD;   // 1024 flattened basis-projection columns

// ---------------------------------------------------------------------------
// Kernel 1: Hb[n][b*256+o] = sum_i X[n][i] * basis[b][i][o]
// One 16x16 fp32 WMMA tile per wave; 8 waves/block -> 32-row x 64-col tile.
// A frag (16x4 f32): lane l -> row l&15, k = k0 + 2*(l>>4) + {0,1}  (float2)
// B frag (4x16 f32): lane l -> col l&15, k = k0 + 2*(l>>4) + {0,1}  (strided)
// C/D (16x16 f32):   vgpr v -> row v + 8*(l>>4), col l&15
// ---------------------------------------------------------------------------
__global__ __launch_bounds__(256) void rgcn_hb_gemm(
    const float* __restrict__ X,
    const float* __restrict__ basis,
    float* __restrict__ Hb)
{
  const int lane = threadIdx.x & 31;
  const int wave = threadIdx.x >> 5;
  const int row0 = blockIdx.x * 32 + (wave >> 2) * 16;
  const int col0 = blockIdx.y * 64 + (wave & 3) * 16;   // flattened over 1024
  const int b    = col0 >> 8;                           // basis index
  const int o0   = col0 & 255;                          // column within basis b
  const int n    = lane & 15;
  const int kh   = lane >> 4;                           // 0 or 1

  const float* __restrict__ A = X + (size_t)(row0 + n) * kD + 2 * kh;
  const float* __restrict__ B = basis + (size_t)b * kD * kD
                                      + (size_t)(2 * kh) * kD + (o0 + n);

  v8f c = {0.f, 0.f, 0.f, 0.f, 0.f, 0.f, 0.f, 0.f};
  #pragma unroll 4
  for (int k = 0; k < kD; k += 4) {
    v2f a = *(const v2f*)(A + k);
    v2f bb;
    bb.x = B[(size_t)k * kD];
    bb.y = B[(size_t)(k + 1) * kD];
    c = __builtin_amdgcn_wmma_f32_16x16x4_f32(false, a, false, bb,
                                              (short)0, c, false, false);
  }

  float* __restrict__ out = Hb + (size_t)(row0 + 8 * kh) * kHbCols + col0 + n;
  #pragma unroll
  for (int v = 0; v < 8; ++v)
    out[(size_t)v * kHbCols] = c[v];
}

// ---------------------------------------------------------------------------
// Kernel 2: per-edge message + scatter-add.
// agg[dst][o] += sum_b comp[etype][b] * Hb[src][b*256+o]
// One wave32 per edge, 8 features/lane, float4 gathers (L2-resident Hb).
// ---------------------------------------------------------------------------
__global__ __launch_bounds__(256) void rgcn_edge_scatter(
    const float* __restrict__ Hb,
    const int* __restrict__ src,
    const int* __restrict__ dst,
    const int* __restrict__ etype,
    const float* __restrict__ comp,
    float* __restrict__ agg)
{
  const int lane = threadIdx.x & 31;
  const int wave = threadIdx.x >> 5;
  const int e = blockIdx.x * 8 + wave;
  if (e >= kEdges) return;

  const int s = __builtin_amdgcn_readfirstlane(src[e]);
  const int d = __builtin_amdgcn_readfirstlane(dst[e]);
  const int r = __builtin_amdgcn_readfirstlane(etype[e]);

  const float c0 = comp[r * kNB + 0];
  const float c1 = comp[r * kNB + 1];
  const float c2 = comp[r * kNB + 2];
  const float c3 = comp[r * kNB + 3];

  const float* __restrict__ h = Hb + (size_t)s * kHbCols + lane * 8;
  float* __restrict__ a       = agg + (size_t)d * kD    + lane * 8;

  #pragma unroll
  for (int half = 0; half < 2; ++half) {
    const int off = half * 4;
    float4 x0 = *(const float4*)(h + 0 * kD + off);
    float4 x1 = *(const float4*)(h + 1 * kD + off);
    float4 x2 = *(const float4*)(h + 2 * kD + off);
    float4 x3 = *(const float4*)(h + 3 * kD + off);
    atomicAdd(a + off + 0, c0 * x0.x + c1 * x1.x + c2 * x2.x + c3 * x3.x);
    atomicAdd(a + off + 1, c0 * x0.y + c1 * x1.y + c2 * x2.y + c3 * x3.y);
    atomicAdd(a + off + 2, c0 * x0.z + c1 * x1.z + c2 * x2.z + c3 * x3.z);
    atomicAdd(a + off + 3, c0 * x0.w + c1 * x1.w + c2 * x2.w + c3 * x3.w);
  }
}

// ---------------------------------------------------------------------------
// Kernel 3: out = LayerNorm(GELU(agg + X@loop_w + bias + X)) * gamma + beta
// 512 threads = 16 waves; block tile = 16 rows x 256 cols (full LN row in
// block). Each wave: one 16x16 fp32 WMMA tile, epilogue to LDS, then wave w
// LayerNorm-reduces tile row w with wave32 shfl_xor.
// ---------------------------------------------------------------------------
__global__ __launch_bounds__(512) void rgcn_final(
    const float* __restrict__ X,
    const float* __restrict__ loop_w,
    const float* __restrict__ agg,
    const float* __restrict__ h_bias,
    const float* __restrict__ gamma,
    const float* __restrict__ beta,
    float* __restrict__ out)
{
  __shared__ float hbuf[16][kD + 8];

  const int lane    = threadIdx.x & 31;
  const int wave    = threadIdx.x >> 5;     // 0..15
  const int rowBase = blockIdx.x * 16;
  const int col0    = wave * 16;
  const int n       = lane & 15;
  const int kh      = lane >> 4;

  const float* __restrict__ A = X + (size_t)(rowBase + n) * kD + 2 * kh;
  const float* __restrict__ B = loop_w + (size_t)(2 * kh) * kD + (col0 + n);

  v8f c = {0.f, 0.f, 0.f, 0.f, 0.f, 0.f, 0.f, 0.f};
  #pragma unroll 4
  for (int k = 0; k < kD; k += 4) {
    v2f a = *(const v2f*)(A + k);
    v2f bb;
    bb.x = B[(size_t)k * kD];
    bb.y = B[(size_t)(k + 1) * kD];
    c = __builtin_amdgcn_wmma_f32_16x16x4_f32(false, a, false, bb,
                                              (short)0, c, false, false);
  }

  // Epilogue: + agg + bias + residual, exact GELU, stash in LDS
  const int col = col0 + n;
  #pragma unroll
  for (int v = 0; v < 8; ++v) {
    const int lr  = v + 8 * kh;             // local row in tile
    const int row = rowBase + lr;
    float t = c[v] + agg[(size_t)row * kD + col]
                   + h_bias[col]
                   + X[(size_t)row * kD + col];
    t = 0.5f * t * (1.0f + erff(t * 0.70710678118654752440f));
    hbuf[lr][col] = t;
  }
  __syncthreads();

  // LayerNorm: wave w owns tile row w; 8 elements per lane.
  float vals[8];
  float s = 0.f, sq = 0.f;
  #pragma unroll
  for (int j = 0; j < 8; ++j) {
    float x = hbuf[wave][lane * 8 + j];
    vals[j] = x;
    s  += x;
    sq += x * x;
  }
  #pragma unroll
  for (int off = 16; off >= 1; off >>= 1) {
    s  += __shfl_xor(s,  off, 32);
    sq += __shfl_xor(sq, off, 32);
  }
  const float mu  = s * (1.0f / 256.0f);
  const float var = sq * (1.0f / 256.0f) - mu * mu;
  const float inv = rsqrtf(var + 1e-5f);

  const int row = rowBase + wave;
  float* __restrict__ o = out + (size_t)row * kD + lane * 8;
  #pragma unroll
  for (int j = 0; j < 8; ++j) {
    const int cidx = lane * 8 + j;
    o[j] = (vals[j] - mu) * inv * gamma[cidx] + beta[cidx];
  }
}

// ---------------------------------------------------------------------------
extern "C" void kernel_launch(void* const* d_in, const int* in_sizes, int n_in,
                              void* d_out, int out_size, void* d_ws, size_t ws_size,
                              hipStream_t stream) {
  (void)in_sizes; (void)n_in; (void)out_size; (void)ws_size;

  const float* X      = (const float*)d_in[0];
  const int*   src    = (const int*)  d_in[1];
  const int*   dst    = (const int*)  d_in[2];
  const int*   etype  = (const int*)  d_in[3];
  const float* basis  = (const float*)d_in[4];
  const float* comp   = (const float*)d_in[5];
  const float* loop_w = (const float*)d_in[6];
  const float* h_bias = (const float*)d_in[7];
  const float* gamma  = (const float*)d_in[8];
  const float* beta   = (const float*)d_in[9];
  float* out = (float*)d_out;

  // Workspace: agg [20000,256] f32, then Hb [20000,1024] f32 (~102.4 MB total)
  float* agg = (float*)d_ws;
  float* Hb  = (float*)((char*)d_ws + (size_t)kNodes * kD * sizeof(float));

  hipMemsetAsync(agg, 0, (size_t)kNodes * kD * sizeof(float), stream);

  dim3 g1(kNodes / 32, kHbCols / 64);                 // 625 x 16 blocks
  rgcn_hb_gemm<<<g1, 256, 0, stream>>>(X, basis, Hb);

  rgcn_edge_scatter<<<kEdges / 8, 256, 0, stream>>>(Hb, src, dst, etype, comp, agg);

  rgcn_final<<<kNodes / 16, 512, 0, stream>>>(X, loop_w, agg, h_bias, gamma, beta, out);
}